// FeaturesLoss_45243185496268
// MI455X (gfx1250) — compile-verified
//
#include <hip/hip_runtime.h>

typedef float v2f __attribute__((ext_vector_type(2)));
typedef float v8f __attribute__((ext_vector_type(8)));

#define N_ROWS 8192
#define DIM    128
#define BT     128     // tile edge per block (128x128 gram tile)
#define LSTRIDE 132    // padded LDS row stride in floats (132%64=4 -> conflict-free, 16B aligned)
#define NCLS   64      // padded class count (actual 50)
#define MARGIN 2.0f
#define EPS_F  1e-9f

// workspace layout:
//   +0   : double pos_sum, double neg_sum
//   +16  : uint hist[NCLS]
//   +272 : float sq[N_ROWS]

__global__ __launch_bounds__(128) void init_ws_kernel(double* acc, unsigned int* hist) {
    int t = threadIdx.x;
    if (t < 2)    acc[t]  = 0.0;
    if (t < NCLS) hist[t] = 0u;
}

__global__ __launch_bounds__(256) void hist_kernel(const int* __restrict__ lbl,
                                                   unsigned int* __restrict__ hist) {
    int i = blockIdx.x * 256 + threadIdx.x;
    if (i < N_ROWS) atomicAdd(&hist[lbl[i] & (NCLS - 1)], 1u);
}

__global__ __launch_bounds__(128) void sq_kernel(const float* __restrict__ X,
                                                 float* __restrict__ sq) {
    int i = blockIdx.x * 128 + threadIdx.x;
    const float4* row = (const float4*)(X + (size_t)i * DIM);
    float s = 0.f;
    #pragma unroll 8
    for (int c = 0; c < DIM / 4; ++c) {
        float4 v = row[c];
        s += v.x * v.x + v.y * v.y + v.z * v.z + v.w * v.w;
    }
    sq[i] = s;
}

__global__ __launch_bounds__(256) void gram_loss_kernel(const float* __restrict__ X,
                                                        const int* __restrict__ lbl,
                                                        const float* __restrict__ sq,
                                                        double* __restrict__ acc) {
    int bi = blockIdx.x, bj = blockIdx.y;
    if (bi > bj) return;  // symmetric: upper triangle only, off-diag weighted x2

    __shared__ float lA[BT * LSTRIDE];
    __shared__ float lB[BT * LSTRIDE];
    __shared__ float sqA[BT], sqB[BT];
    __shared__ int   labA[BT], labB[BT];
    __shared__ float red[256];

    int tid = threadIdx.x;
    int rowA0 = bi * BT, rowB0 = bj * BT;

    // Stage both 128x128 fp32 panels into LDS (b128 global loads / b128 LDS stores)
    for (int idx = tid; idx < BT * (DIM / 4); idx += 256) {
        int row = idx >> 5;        // DIM/4 == 32 float4 per row
        int c4  = idx & 31;
        float4 va = *((const float4*)(X + (size_t)(rowA0 + row) * DIM) + c4);
        float4 vb = *((const float4*)(X + (size_t)(rowB0 + row) * DIM) + c4);
        *(float4*)(&lA[row * LSTRIDE + c4 * 4]) = va;
        *(float4*)(&lB[row * LSTRIDE + c4 * 4]) = vb;
    }
    if (tid < BT) {
        sqA[tid]  = sq[rowA0 + tid];
        sqB[tid]  = sq[rowB0 + tid];
        labA[tid] = lbl[rowA0 + tid];
        labB[tid] = lbl[rowB0 + tid];
    }
    __syncthreads();

    // 8 waves; wave owns a 32-row x 64-col sub-tile: 2 A-frags x 4 B-frags -> 8 WMMA / 6 LDS loads
    int wave  = tid >> 5;
    int lane  = tid & 31;
    int rbase = (wave & 3) * 32;       // row strip of 32
    int cbase = (wave >> 2) * 64;      // column half of 64
    int m     = lane & 15;             // row within 16-row fragment
    int kp    = (lane >> 4) << 1;      // K-pair: lanes 0-15 -> K{0,1}, lanes 16-31 -> K{2,3}

    v8f c00 = {}, c01 = {}, c02 = {}, c03 = {};
    v8f c10 = {}, c11 = {}, c12 = {}, c13 = {};

    const float* aRow0 = &lA[(rbase +  0 + m) * LSTRIDE + kp];
    const float* aRow1 = &lA[(rbase + 16 + m) * LSTRIDE + kp];
    const float* bRow0 = &lB[(cbase +  0 + m) * LSTRIDE + kp];
    const float* bRow1 = &lB[(cbase + 16 + m) * LSTRIDE + kp];
    const float* bRow2 = &lB[(cbase + 32 + m) * LSTRIDE + kp];
    const float* bRow3 = &lB[(cbase + 48 + m) * LSTRIDE + kp];

    #pragma unroll 4
    for (int k0 = 0; k0 < DIM; k0 += 4) {
        v2f a0 = *(const v2f*)(aRow0 + k0);
        v2f a1 = *(const v2f*)(aRow1 + k0);
        v2f b0 = *(const v2f*)(bRow0 + k0);
        v2f b1 = *(const v2f*)(bRow1 + k0);
        v2f b2 = *(const v2f*)(bRow2 + k0);
        v2f b3 = *(const v2f*)(bRow3 + k0);
        c00 = __builtin_amdgcn_wmma_f32_16x16x4_f32(false, a0, false, b0, (short)0, c00, false, false);
        c01 = __builtin_amdgcn_wmma_f32_16x16x4_f32(false, a0, false, b1, (short)0, c01, false, false);
        c02 = __builtin_amdgcn_wmma_f32_16x16x4_f32(false, a0, false, b2, (short)0, c02, false, false);
        c03 = __builtin_amdgcn_wmma_f32_16x16x4_f32(false, a0, false, b3, (short)0, c03, false, false);
        c10 = __builtin_amdgcn_wmma_f32_16x16x4_f32(false, a1, false, b0, (short)0, c10, false, false);
        c11 = __builtin_amdgcn_wmma_f32_16x16x4_f32(false, a1, false, b1, (short)0, c11, false, false);
        c12 = __builtin_amdgcn_wmma_f32_16x16x4_f32(false, a1, false, b2, (short)0, c12, false, false);
        c13 = __builtin_amdgcn_wmma_f32_16x16x4_f32(false, a1, false, b3, (short)0, c13, false, false);
    }

    // Fused branchless loss epilogue.
    // C/D layout: VGPR r holds M=r (lanes 0-15) and M=r+8 (lanes 16-31), N = lane&15
    float wgt = (bi == bj) ? 1.0f : 2.0f;
    float posAcc = 0.f, negAcc = 0.f;
    int laneHi = (lane < 16) ? 0 : 8;
    int jLane  = lane & 15;
    #pragma unroll
    for (int s = 0; s < 2; ++s) {
        int iBase = rbase + s * 16 + laneHi;
        #pragma unroll
        for (int jt = 0; jt < 4; ++jt) {
            v8f cc = (s == 0) ? ((jt == 0) ? c00 : (jt == 1) ? c01 : (jt == 2) ? c02 : c03)
                              : ((jt == 0) ? c10 : (jt == 1) ? c11 : (jt == 2) ? c12 : c13);
            int   jloc = cbase + jt * 16 + jLane;
            float sqj  = sqB[jloc];
            int   lj   = labB[jloc];
            #pragma unroll
            for (int r = 0; r < 8; ++r) {
                int   iloc = iBase + r;
                float d2   = fmaxf(sqA[iloc] + sqj - 2.0f * cc[r], 0.0f);
                float h    = fmaxf(MARGIN - sqrtf(d2 + EPS_F), 0.0f);
                bool  pos  = (labA[iloc] == lj);
                posAcc += pos ? d2 : 0.0f;        // v_cndmask, no divergence
                negAcc += pos ? 0.0f : h * h;
            }
        }
    }
    posAcc *= wgt; negAcc *= wgt;

    // block reduction -> one f64 atomic per sum per block
    red[tid] = posAcc;
    __syncthreads();
    for (int s = 128; s > 0; s >>= 1) { if (tid < s) red[tid] += red[tid + s]; __syncthreads(); }
    float blockPos = red[0];
    __syncthreads();
    red[tid] = negAcc;
    __syncthreads();
    for (int s = 128; s > 0; s >>= 1) { if (tid < s) red[tid] += red[tid + s]; __syncthreads(); }
    if (tid == 0) {
        atomicAdd(&acc[0], (double)blockPos);
        atomicAdd(&acc[1], (double)red[0]);
    }
}

__global__ void finalize_kernel(const double* __restrict__ acc,
                                const unsigned int* __restrict__ hist,
                                float* __restrict__ out) {
    if (threadIdx.x == 0 && blockIdx.x == 0) {
        long long npos = 0;
        for (int c = 0; c < NCLS; ++c) {
            long long h = (long long)hist[c];
            npos += h * h;
        }
        long long total = (long long)N_ROWS * (long long)N_ROWS;
        long long nneg  = total - npos;
        double pt = (npos > 0) ? 0.5 * acc[0] / (double)npos : 0.0;
        double nt = (nneg > 0) ? 0.5 * acc[1] / (double)nneg : 0.0;
        out[0] = (float)(pt + nt);
    }
}

extern "C" void kernel_launch(void* const* d_in, const int* in_sizes, int n_in,
                              void* d_out, int out_size, void* d_ws, size_t ws_size,
                              hipStream_t stream) {
    const float* X   = (const float*)d_in[0];
    const int*   lbl = (const int*)d_in[1];
    char* ws = (char*)d_ws;
    double*       acc  = (double*)ws;
    unsigned int* hist = (unsigned int*)(ws + 16);
    float*        sq   = (float*)(ws + 16 + NCLS * sizeof(unsigned int));
    float*        out  = (float*)d_out;

    init_ws_kernel<<<1, 128, 0, stream>>>(acc, hist);
    hist_kernel<<<(N_ROWS + 255) / 256, 256, 0, stream>>>(lbl, hist);
    sq_kernel<<<N_ROWS / 128, 128, 0, stream>>>(X, sq);
    dim3 grid(N_ROWS / BT, N_ROWS / BT);
    gram_loss_kernel<<<grid, 256, 0, stream>>>(X, lbl, sq, acc);
    finalize_kernel<<<1, 64, 0, stream>>>(acc, hist, out);
}